// MRSATSPMConv_46359876993096
// MI455X (gfx1250) — compile-verified
//
#include <hip/hip_runtime.h>

typedef float v2f __attribute__((ext_vector_type(2)));
typedef float v8f __attribute__((ext_vector_type(8)));

#define NNODES 100000
#define NEDGES 600000
#define DIM 128

#define XS_STRIDE 132
#define WS_STRIDE 132
#define AS_STRIDE 132
#define WT_STRIDE 137

// ---------------- Phase A: Y_r = x @ W_r^T (r = rel0, rel1, self->agg) --------
// 256 thr (8 waves) per block, 128 node rows per block. Each wave owns one
// 16-row M-tile and sweeps 8 N-tiles x 32 K-steps of V_WMMA_F32_16X16X4_F32.
__global__ void mrsat_transform_kernel(const float* __restrict__ x,
                                       const float* __restrict__ Wrel,
                                       const float* __restrict__ Wself,
                                       float* __restrict__ Y0,
                                       float* __restrict__ Y1,
                                       float* __restrict__ agg) {
  extern __shared__ float smem[];
  float* xs = smem;                     // [128][XS_STRIDE]
  float* ws = smem + 128 * XS_STRIDE;   // [128][WS_STRIDE]

  const int tid  = threadIdx.x;
  const int wave = tid >> 5;
  const int lane = tid & 31;
  const int l    = lane & 15;
  const int half = lane >> 4;
  const int rows0 = blockIdx.x * 128;

  // stage 128x128 x-tile (zero-padded past N)
  #pragma unroll
  for (int j = 0; j < 16; ++j) {
    int fidx = (tid + j * 256) * 4;
    int row = fidx >> 7, col = fidx & 127;
    int g = rows0 + row;
    float4 v = make_float4(0.f, 0.f, 0.f, 0.f);
    if (g < NNODES) v = *(const float4*)(x + g * DIM + col);
    *(float4*)(xs + row * XS_STRIDE + col) = v;
  }

  for (int r = 0; r < 3; ++r) {
    const float* W = (r == 0) ? Wrel : (r == 1) ? (Wrel + DIM * DIM) : Wself;
    float* Y = (r == 0) ? Y0 : (r == 1) ? Y1 : agg;
    __syncthreads();
    #pragma unroll
    for (int j = 0; j < 16; ++j) {
      int fidx = (tid + j * 256) * 4;
      int row = fidx >> 7, col = fidx & 127;
      *(float4*)(ws + row * WS_STRIDE + col) = *(const float4*)(W + fidx);
    }
    __syncthreads();

    const int mrow = wave * 16 + l;
    for (int nt = 0; nt < 8; ++nt) {
      v8f c = {};
      #pragma unroll
      for (int ks = 0; ks < 32; ++ks) {
        int k = ks * 4 + half * 2;                       // A lanes0-15: K=k,k+1; lanes16-31: K=k+2,k+3
        v2f a = *(const v2f*)(xs + mrow * XS_STRIDE + k);
        // B[k][n] = W^T[k][n] = W[n][k]  (W row-major [out][in])
        v2f b = *(const v2f*)(ws + (nt * 16 + l) * WS_STRIDE + k);
        c = __builtin_amdgcn_wmma_f32_16x16x4_f32(false, a, false, b,
                                                  (short)0, c, false, false);
      }
      int gr = rows0 + wave * 16 + half * 8;             // C VGPR v -> row v + 8*half
      #pragma unroll
      for (int v = 0; v < 8; ++v) {
        int g = gr + v;
        if (g < NNODES) Y[g * DIM + nt * 16 + l] = c[v];
      }
    }
  }
}

// ---------------- Phase B: agg[tgt] += Y[src]  (one wave per edge) ------------
__global__ void mrsat_scatter_kernel(const long long* __restrict__ ei,
                                     const float* __restrict__ Y,
                                     float* __restrict__ agg) {
  int wid  = (blockIdx.x * blockDim.x + threadIdx.x) >> 5;
  int lane = threadIdx.x & 31;
  if (wid >= NEDGES) return;
  long long s = ei[wid];            // row 0: src
  long long t = ei[NEDGES + wid];   // row 1: tgt
  float4 v = *(const float4*)(Y + s * DIM + lane * 4);
  float* a = agg + t * DIM + lane * 4;
  atomicAdd(a + 0, v.x);
  atomicAdd(a + 1, v.y);
  atomicAdd(a + 2, v.z);
  atomicAdd(a + 3, v.w);
}

// ---------------- Phase C: Conv1d(k=3, pad=1) over node axis + bias + ReLU ----
// out[n,o] = relu(b[o] + sum_t sum_i agg[n-1+t, i] * conv_w[o, i, t])
// 64 nodes/block; halo rows in LDS; one weight tap staged per iteration,
// accumulators persist across taps.
__global__ void mrsat_conv_kernel(const float* __restrict__ agg,
                                  const float* __restrict__ conv_w,
                                  const float* __restrict__ conv_b,
                                  float* __restrict__ out) {
  extern __shared__ float smem[];
  float* as = smem;                   // [66][AS_STRIDE]: rows n0-1 .. n0+64
  float* wt = smem + 66 * AS_STRIDE;  // [128][WT_STRIDE]: Wt[i][o] for current tap

  const int tid  = threadIdx.x;
  const int wave = tid >> 5;
  const int lane = tid & 31;
  const int l    = lane & 15;
  const int half = lane >> 4;
  const int rows0 = blockIdx.x * 64;

  for (int idx = tid; idx < 66 * DIM; idx += 256) {
    int row = idx >> 7, col = idx & 127;
    int g = rows0 + row - 1;
    as[row * AS_STRIDE + col] = (g >= 0 && g < NNODES) ? agg[g * DIM + col] : 0.0f;
  }

  v8f acc[4] = {};
  const int colg = wave * 16 + l;     // this wave's N-tile column

  for (int t = 0; t < 3; ++t) {
    __syncthreads();                  // covers as-staging at t=0, wt reuse later
    for (int idx = tid; idx < DIM * DIM; idx += 256) {
      int i = idx & 127, o = idx >> 7;        // coalesced-ish stride-3 global reads
      wt[i * WT_STRIDE + o] = conv_w[o * (DIM * 3) + i * 3 + t];
    }
    __syncthreads();
    #pragma unroll
    for (int ks = 0; ks < 32; ++ks) {
      int k = ks * 4 + half * 2;
      v2f b;
      b.x = wt[k * WT_STRIDE + colg];
      b.y = wt[(k + 1) * WT_STRIDE + colg];
      #pragma unroll
      for (int mt = 0; mt < 4; ++mt) {
        v2f a = *(const v2f*)(as + (mt * 16 + l + t) * AS_STRIDE + k);
        acc[mt] = __builtin_amdgcn_wmma_f32_16x16x4_f32(false, a, false, b,
                                                        (short)0, acc[mt], false, false);
      }
    }
  }

  float bias = conv_b[colg];
  #pragma unroll
  for (int mt = 0; mt < 4; ++mt) {
    int gr = rows0 + mt * 16 + half * 8;
    #pragma unroll
    for (int v = 0; v < 8; ++v) {
      int g = gr + v;
      if (g < NNODES) out[g * DIM + colg] = fmaxf(acc[mt][v] + bias, 0.0f);
    }
  }
}

extern "C" void kernel_launch(void* const* d_in, const int* in_sizes, int n_in,
                              void* d_out, int out_size, void* d_ws, size_t ws_size,
                              hipStream_t stream) {
  const float*     x      = (const float*)d_in[0];
  const long long* ei0    = (const long long*)d_in[1];
  const long long* ei1    = (const long long*)d_in[2];
  const float*     Wrel   = (const float*)d_in[3];  // [2][128][128]
  const float*     Wself  = (const float*)d_in[4];  // [128][128]
  const float*     conv_w = (const float*)d_in[5];  // [128][128][3]
  const float*     conv_b = (const float*)d_in[6];  // [128]
  float* out = (float*)d_out;

  float* Y0  = (float*)d_ws;                         // [N][128]
  float* Y1  = Y0 + (size_t)NNODES * DIM;            // [N][128]
  float* agg = Y1 + (size_t)NNODES * DIM;            // [N][128] (Yself + scatter)

  size_t lds_a = (size_t)(128 * XS_STRIDE + 128 * WS_STRIDE) * sizeof(float);
  mrsat_transform_kernel<<<(NNODES + 127) / 128, 256, lds_a, stream>>>(
      x, Wrel, Wself, Y0, Y1, agg);

  int sblocks = (NEDGES * 32 + 255) / 256;
  mrsat_scatter_kernel<<<sblocks, 256, 0, stream>>>(ei0, Y0, agg);
  mrsat_scatter_kernel<<<sblocks, 256, 0, stream>>>(ei1, Y1, agg);

  size_t lds_c = (size_t)(66 * AS_STRIDE + 128 * WT_STRIDE) * sizeof(float);
  mrsat_conv_kernel<<<(NNODES + 63) / 64, 256, lds_c, stream>>>(
      agg, conv_w, conv_b, out);
}